// YOLOLayer_1726576856077
// MI455X (gfx1250) — compile-verified
//
#include <hip/hip_runtime.h>

// ---------------- problem constants ----------------
#define NB      16            // batch
#define NA      3             // anchors
#define NG      76            // grid h == w
#define NT      50            // targets per image
#define NCH     85            // 5 + 80 classes
#define NCELL2D (NG * NG)     // 5776
#define NCELLS  (NB * NA * NCELL2D)   // 277248
#define THREADS 256
#define MBLOCKS 256

typedef float v2f __attribute__((ext_vector_type(2)));
typedef float v8f __attribute__((ext_vector_type(8)));

__device__ __forceinline__ float sigf(float v) {
    return 1.0f / (1.0f + __expf(-v));
}

// ---------------- init workspace ----------------
__global__ void init_k(int* __restrict__ maskIdx, int* __restrict__ cellzero) {
    int i = blockIdx.x * blockDim.x + threadIdx.x;
    if (i < NCELLS) maskIdx[i] = -1;
    if (i < NCELL2D) cellzero[i] = 0;
}

// ---------------- build targets (1 wave) ----------------
__global__ void build_targets_k(const float* __restrict__ targets,
                                const float* __restrict__ anchors,
                                int* __restrict__ maskIdx,
                                int* __restrict__ cellzero,
                                float4* __restrict__ recs) {
    int b = threadIdx.x;
    if (b >= NB) return;
    const float inv_stride = (float)NG / 608.0f;   // 1/8
    float aw[NA], ah[NA];
    for (int a = 0; a < NA; ++a) {
        aw[a] = anchors[2 * a]     * inv_stride;
        ah[a] = anchors[2 * a + 1] * inv_stride;
    }
    const float* tb = targets + b * NT * 5;
    const float cx0 = tb[1], cy0 = tb[2], cw0 = tb[3], ch0 = tb[4];
    for (int t = 0; t < NT; ++t) {
        float lab = tb[t*5+0], cx = tb[t*5+1], cy = tb[t*5+2],
              cw  = tb[t*5+3], ch = tb[t*5+4];
        bool mm = (lab + cx + cy + cw + ch) > 0.0f;
        float gx = (mm ? cx : cx0) * (float)NG;
        float gy = (mm ? cy : cy0) * (float)NG;
        float gw = (mm ? cw : cw0) * (float)NG;
        float gh = (mm ? ch : ch0) * (float)NG;
        int gi = (int)gx, gj = (int)gy;
        float best = -1.0f; int bn = 0; bool ign = false;
        for (int a = 0; a < NA; ++a) {
            float inter = fmaxf(fminf(gw, aw[a]) + 1.0f, 0.0f) *
                          fmaxf(fminf(gh, ah[a]) + 1.0f, 0.0f);
            float den = (gw + 1.0f) * (gh + 1.0f) +
                        (aw[a] + 1.0f) * (ah[a] + 1.0f) - inter + 1e-12f;
            float iou = inter / den;
            ign |= (iou > 0.5f);
            if (iou > best) { best = iou; bn = a; }   // first-max == jnp.argmax
        }
        if (ign) atomicOr(&cellzero[gj * NG + gi], 1);
        int rid = b * NT + t;
        recs[rid] = make_float4(gx - (float)gi, gy - (float)gj,
                                __logf(gw / aw[bn] + 1e-16f),
                                __logf(gh / ah[bn] + 1e-16f));
        int cell = ((b * NA + bn) * NG + gj) * NG + gi;
        maskIdx[cell] = rid;   // sequential per batch -> deterministic last wins
    }
}

// ---------------- main sweep + WMMA block reduction ----------------
__global__ void __launch_bounds__(THREADS)
yolo_main_k(const float* __restrict__ sample,
            const int* __restrict__ maskIdx,
            const int* __restrict__ cellzero,
            const float4* __restrict__ recs,
            float* __restrict__ blockSums) {
    float acc[8] = {0.f,0.f,0.f,0.f,0.f,0.f,0.f,0.f};

    const int gstride = gridDim.x * blockDim.x;
    for (int cell = blockIdx.x * blockDim.x + threadIdx.x; cell < NCELLS;
         cell += gstride) {
        int ji = cell % NCELL2D;         // j*NG + i  (shared ignore grid index)
        int ba = cell / NCELL2D;         // b*NA + a
        const float* sp = sample + (size_t)ba * (NCH * NCELL2D) + ji;
        // only 5 of 85 channels are live
        float p0 = sp[0];
        float p1 = sp[1 * NCELL2D];
        float p2 = sp[2 * NCELL2D];
        float p3 = sp[3 * NCELL2D];
        float p4 = sp[4 * NCELL2D];

        int id = maskIdx[cell];
        float m = (id >= 0) ? 1.0f : 0.0f;
        float4 tt = (id >= 0) ? recs[id] : make_float4(0.f, 0.f, 0.f, 0.f);
        float cz  = cellzero[ji] ? 0.0f : 1.0f;
        float cmf = (id >= 0) ? 0.0f : cz;      // conf_mask * (1-mask)

        float x = sigf(p0), y = sigf(p1), pc = sigf(p4);
        float dx = x  - tt.x;
        float dy = y  - tt.y;
        float dw = p2 - tt.z;
        float dh = p3 - tt.w;

        acc[0] += m;
        acc[1] += m * dx * dx;
        acc[2] += m * dy * dy;
        acc[3] += m * dw * dw;
        acc[4] += m * dh * dh;
        acc[5] += m   * (-__logf(pc + 1e-12f));          // obj BCE (t=1)
        acc[6] += cmf * (-__logf(1.0f - pc + 1e-12f));   // noobj BCE (t=0)
        acc[7] += cmf;
    }

    // ---- cross-lane reduction via V_WMMA_F32_16X16X4_F32 (B = ones) ----
    __shared__ float red[THREADS / 32][32][8];   // 8 KB
    __shared__ float wsum[THREADS / 32][8];
    const int lane = threadIdx.x & 31;
    const int wave = threadIdx.x >> 5;
    for (int c = 0; c < 8; ++c) red[wave][lane][c] = acc[c];
    __syncthreads();

    // A(16x4) layout: lane L -> row M=L%16; K = {0,1} for L<16, {2,3} for L>=16.
    // Row m carries component (m%8), half (m>>3); 4 chained WMMAs consume all
    // 32 lanes x 8 components, accumulating row sums in D (all N columns equal).
    const int m16   = lane & 15;
    const int comp  = m16 & 7;
    const int half  = m16 >> 3;
    const int kbase = (lane < 16) ? 0 : 2;
    v2f bones; bones.x = 1.0f; bones.y = 1.0f;
    v8f d = {0.f,0.f,0.f,0.f,0.f,0.f,0.f,0.f};
    #pragma unroll
    for (int s = 0; s < 4; ++s) {
        int src = half * 16 + s * 4 + kbase;
        v2f a;
        a.x = red[wave][src + 0][comp];
        a.y = red[wave][src + 1][comp];
        d = __builtin_amdgcn_wmma_f32_16x16x4_f32(
                false, a, false, bones, (short)0, d, false, false);
    }
    // D vgpr r: lanes 0-15 hold row r, lanes 16-31 hold row r+8.
    // total(comp c) = D[c][*] + D[c+8][*]  ->  d[c] + shfl_xor16(d[c]) at lane 0.
    if (lane == 0 || lane == 16) {
        // lane 0 combines with lane 16 via xor shuffle below
    }
    #pragma unroll
    for (int c = 0; c < 8; ++c) {
        float other = __shfl(d[c], lane ^ 16, 32);
        float tot = d[c] + other;
        if (lane == 0) wsum[wave][c] = tot;
    }
    __syncthreads();
    if (threadIdx.x < 8) {
        float s = 0.0f;
        for (int w = 0; w < THREADS / 32; ++w) s += wsum[w][threadIdx.x];
        blockSums[blockIdx.x * 8 + threadIdx.x] = s;
    }
}

// ---------------- finalize: 7 outputs ----------------
__global__ void finalize_k(const float* __restrict__ blockSums,
                           float* __restrict__ out) {
    if (threadIdx.x == 0 && blockIdx.x == 0) {
        float s[8] = {0.f,0.f,0.f,0.f,0.f,0.f,0.f,0.f};
        for (int b = 0; b < MBLOCKS; ++b)
            for (int c = 0; c < 8; ++c) s[c] += blockSums[b * 8 + c];
        float n  = fmaxf(s[0], 1.0f);
        float lx = 2.0f * s[1] / n;
        float ly = 2.0f * s[2] / n;
        float lw = 1.6f * s[3] / n;
        float lh = 1.6f * s[4] / n;
        float lobj   = s[5] / n;
        float lnoobj = 0.5f * s[6] / fmaxf(s[7], 1.0f);
        out[0] = lx + ly + lw + lh + lnoobj + lobj;
        out[1] = lx; out[2] = ly; out[3] = lw; out[4] = lh;
        out[5] = lobj; out[6] = lnoobj;
    }
}

// ---------------- host entry ----------------
extern "C" void kernel_launch(void* const* d_in, const int* in_sizes, int n_in,
                              void* d_out, int out_size, void* d_ws, size_t ws_size,
                              hipStream_t stream) {
    const float* sample  = (const float*)d_in[0];
    const float* targets = (const float*)d_in[1];
    const float* anchors = (const float*)d_in[2];
    float* out = (float*)d_out;

    // workspace layout (all offsets 16B aligned)
    int*    maskIdx   = (int*)d_ws;                          // NCELLS ints
    int*    cellzero  = maskIdx + NCELLS;                    // 5776 ints
    float*  blockSums = (float*)(cellzero + NCELL2D);        // MBLOCKS*8 floats
    float4* recs      = (float4*)(blockSums + MBLOCKS * 8);  // NB*NT float4

    init_k<<<(NCELLS + THREADS - 1) / THREADS, THREADS, 0, stream>>>(maskIdx, cellzero);
    build_targets_k<<<1, 32, 0, stream>>>(targets, anchors, maskIdx, cellzero, recs);
    yolo_main_k<<<MBLOCKS, THREADS, 0, stream>>>(sample, maskIdx, cellzero, recs, blockSums);
    finalize_k<<<1, 32, 0, stream>>>(blockSums, out);
}